// UpSampling_352187318661
// MI455X (gfx1250) — compile-verified
//
#include <hip/hip_runtime.h>

// ---------------------------------------------------------------------------
// StyleGAN 2x upsample (zero-insert + [1,4,6,4,1]/8 FIR), separable D/H/W.
// Per-axis stencil:
//   out[2i]   = 0.5*x[i-1] + 0.5*x[i]
//   out[2i+1] = 0.125*x[i-1] + 0.75*x[i] + 0.125*x[i+1]   (zero-padded edges)
// Each 1-D pass: D(16x16) = M(16x12 banded) * X(12x16) via three chained
// V_WMMA_F32_16X16X4_F32. Input staged with GLOBAL_LOAD_ASYNC_TO_LDS_B32,
// output streamed with GLOBAL_STORE_ASYNC_FROM_LDS_B128 (coalesced 16B).
// ---------------------------------------------------------------------------

typedef __attribute__((ext_vector_type(2))) float v2f;
typedef __attribute__((ext_vector_type(8))) float v8f;

#define DI 48          // input spatial extent (cubic)
#define DO 96          // output spatial extent
#define NC 128         // N*C = 4*32 channel slabs
#define TI 8           // input tile per axis
#define THL 10         // halo'd input tile per axis
#define TP 12          // padded axis length (slots 10,11 held at zero)
#define TO 16          // output tile per axis

// Banded upsample matrix M[j][i]; input slot i corresponds to x[base-1+i].
__device__ __forceinline__ float mcoef(int j, int i) {
  int a = j >> 1;
  if (j & 1) {
    if (i == a)     return 0.125f;
    if (i == a + 1) return 0.75f;
    if (i == a + 2) return 0.125f;
  } else {
    if (i == a || i == a + 1) return 0.5f;
  }
  return 0.0f;
}

__device__ __forceinline__ unsigned lds_off(const void* p) {
  return (unsigned)(uintptr_t)p;   // low 32 bits of flat shared addr = LDS offset
}

__device__ __forceinline__ void async_load_b32(unsigned lds_addr, unsigned byte_off,
                                               const void* sbase) {
  asm volatile("global_load_async_to_lds_b32 %0, %1, %2"
               :: "v"(lds_addr), "v"(byte_off), "s"(sbase) : "memory");
}

__device__ __forceinline__ void async_store_b128(unsigned byte_off, unsigned lds_addr,
                                                 const void* sbase) {
  asm volatile("global_store_async_from_lds_b128 %0, %1, %2"
               :: "v"(byte_off), "v"(lds_addr), "s"(sbase) : "memory");
}

__device__ __forceinline__ void wait_async0() {
  asm volatile("s_wait_asynccnt 0x0" ::: "memory");
}

__global__ __launch_bounds__(256)
void upfirdn3d_x2_wmma(const float* __restrict__ x, float* __restrict__ out) {
  __shared__ __align__(16) float sIn[THL * THL * TP];  // 1200 : (d,h)x(w pad12)
  __shared__ __align__(16) float sT1[THL * TP * TO];   // 1920 : (d)(h pad12)(w2)
  __shared__ __align__(16) float sT2[TP * TO * TO];    // 3072 : (d pad12)(h2)(w2)
  __shared__ __align__(16) float sT3[TO * TO * TO];    // 4096 : output tile

  const int tid  = threadIdx.x;
  const int lane = tid & 31;
  const int wave = tid >> 5;
  const int col  = lane & 15;   // N index (row id within a 16-row group)
  const int hi   = lane >> 4;   // half-wave selector for K / M mapping

  const int tile = blockIdx.x;  // 0..215 (6x6x6 tiles)
  const int nc   = blockIdx.y;  // 0..127
  const int tx = tile % 6;
  const int ty = (tile / 6) % 6;
  const int tz = tile / 36;

  const float* src = x + (size_t)nc * (DI * DI * DI);
  float*       dst = out + (size_t)nc * (DO * DO * DO);

  // ---- zero pad slots (processed-axis indices 10,11 of each buffer) ----
  for (int p = tid; p < THL * THL * 2; p += 256) {      // sIn w=10,11
    int dh = p >> 1;
    sIn[dh * TP + 10 + (p & 1)] = 0.0f;
  }
  for (int p = tid; p < THL * 2 * TO; p += 256) {       // sT1 h=10,11
    int d = p / (2 * TO);
    int r = p % (2 * TO);
    sT1[(d * TP + 10 + (r >> 4)) * TO + (r & 15)] = 0.0f;
  }
  for (int p = tid; p < 2 * TO * TO; p += 256) {        // sT2 d=10,11
    sT2[10 * TO * TO + p] = 0.0f;
  }

  // ---- async-fill halo'd 10^3 input tile (zero at volume boundaries) ----
  for (int idx = tid; idx < THL * THL * THL; idx += 256) {
    int lw = idx % THL;
    int lh = (idx / THL) % THL;
    int ld = idx / (THL * THL);
    int gw = tx * TI + lw - 1;
    int gh = ty * TI + lh - 1;
    int gd = tz * TI + ld - 1;
    int li = (ld * THL + lh) * TP + lw;
    if ((unsigned)gw < DI && (unsigned)gh < DI && (unsigned)gd < DI) {
      unsigned gbyte = 4u * (unsigned)((gd * DI + gh) * DI + gw);
      async_load_b32(lds_off(&sIn[li]), gbyte, src);
    } else {
      sIn[li] = 0.0f;
    }
  }
  wait_async0();

  // ---- constant A fragments (ISA 16x4 f32 A layout: row=lane&15,
  //      VGPR0 -> K=2*hi, VGPR1 -> K=2*hi+1), one per K-chunk ----
  v2f afr[3];
  {
    int j = col;
    #pragma unroll
    for (int c = 0; c < 3; ++c) {
      int i0 = 4 * c + 2 * hi;
      afr[c].x = mcoef(j, i0);
      afr[c].y = mcoef(j, i0 + 1);
    }
  }

  __syncthreads();

  // ---- pass 1 (W axis): rows r = d*10+h (0..99), X[i][r] = sIn[r*12+i] ----
  auto pass1 = [&](int g) {
    int r  = g * 16 + col;
    int rc = (r < 100) ? r : 99;          // clamp for safe B reads
    v8f acc = {};
    #pragma unroll
    for (int c = 0; c < 3; ++c) {
      int i0 = 4 * c + 2 * hi;
      v2f b = *(const v2f*)&sIn[rc * TP + i0];   // single ds_load_b64
      acc = __builtin_amdgcn_wmma_f32_16x16x4_f32(
          false, afr[c], false, b, (short)0, acc, false, false);
    }
    if (r < 100) {
      int d = r / 10, h = r % 10;
      #pragma unroll
      for (int n = 0; n < 8; ++n)
        sT1[(d * TP + h) * TO + (n + 8 * hi)] = acc[n];
    }
  };
  if (wave < 7) pass1(wave);
  __syncthreads();

  // ---- pass 2 (H axis): rows r = d*16+w2 (0..159),
  //      X[i][r] = sT1[(d*12+i)*16 + w2] ; dst sT2[d*256 + j*16 + w2] ----
  auto pass2 = [&](int g) {
    int r  = g * 16 + col;
    int d  = r >> 4;
    int w2 = r & 15;
    v8f acc = {};
    #pragma unroll
    for (int c = 0; c < 3; ++c) {
      int i0 = 4 * c + 2 * hi;
      v2f b;
      b.x = sT1[(d * TP + i0    ) * TO + w2];
      b.y = sT1[(d * TP + i0 + 1) * TO + w2];
      acc = __builtin_amdgcn_wmma_f32_16x16x4_f32(
          false, afr[c], false, b, (short)0, acc, false, false);
    }
    #pragma unroll
    for (int n = 0; n < 8; ++n)
      sT2[d * 256 + (n + 8 * hi) * TO + w2] = acc[n];
  };
  pass2(wave);
  if (wave < 2) pass2(wave + 8);
  __syncthreads();

  // ---- pass 3 (D axis): rows r = h2*16+w2 (0..255), X[i][r] = sT2[i*256+r],
  //      stage result tile in sT3[(j)*256 + r] ----
  auto pass3 = [&](int g) {
    int r = g * 16 + col;
    v8f acc = {};
    #pragma unroll
    for (int c = 0; c < 3; ++c) {
      int i0 = 4 * c + 2 * hi;
      v2f b;
      b.x = sT2[(i0    ) * 256 + r];
      b.y = sT2[(i0 + 1) * 256 + r];
      acc = __builtin_amdgcn_wmma_f32_16x16x4_f32(
          false, afr[c], false, b, (short)0, acc, false, false);
    }
    #pragma unroll
    for (int n = 0; n < 8; ++n)
      sT3[(n + 8 * hi) * 256 + r] = acc[n];
  };
  pass3(wave);
  pass3(wave + 8);
  __syncthreads();

  // ---- stream output tile: 1024 async b128 stores, lane-contiguous 16B ----
  #pragma unroll
  for (int k = 0; k < 4; ++k) {
    int e  = (tid + k * 256) * 4;           // element index in 16^3 tile
    int d2 = e >> 8;
    int h2 = (e >> 4) & 15;
    int w2 = e & 15;                        // 0,4,8,12
    unsigned gbyte = 4u * (unsigned)(((tz * TO + d2) * DO + ty * TO + h2) * DO
                                     + tx * TO + w2);
    async_store_b128(gbyte, lds_off(&sT3[e]), dst);
  }
  wait_async0();
}

extern "C" void kernel_launch(void* const* d_in, const int* in_sizes, int n_in,
                              void* d_out, int out_size, void* d_ws, size_t ws_size,
                              hipStream_t stream) {
  (void)in_sizes; (void)n_in; (void)d_ws; (void)ws_size; (void)out_size;
  const float* x = (const float*)d_in[0];
  float* out = (float*)d_out;
  dim3 grid(216, NC);   // 6*6*6 tiles x (N*C) channel slabs
  dim3 block(256);      // 8 wave32
  upfirdn3d_x2_wmma<<<grid, block, 0, stream>>>(x, out);
}